// CustomLoss_32057635897752
// MI455X (gfx1250) — compile-verified
//
#include <hip/hip_runtime.h>

typedef __attribute__((ext_vector_type(2))) float v2f;
typedef __attribute__((ext_vector_type(8))) float v8f;

#define THREADS 256
#define BLOCKS  1024
#define WAVES_PER_BLOCK (THREADS / 32)

// Full 32-lane fp32 sum using V_WMMA_F32_16X16X4_F32 against a ones matrix.
// Pass 1: A vgpr0 = acc, A vgpr1 = 0  ->  D[m][n] = acc[m] + acc[m+16] = s[m]
//         per-lane sum of the 8 D vgprs -> lanes 0-15: sum(s[0..7]),
//                                          lanes 16-31: sum(s[8..15])
// Pass 2: same trick -> every lane holds sum of all 32 partials.
__device__ __forceinline__ float wave_reduce_wmma(float acc) {
    v2f ones; ones[0] = 1.0f; ones[1] = 1.0f;

    v2f a; a[0] = acc; a[1] = 0.0f;
    v8f c = {};
    c = __builtin_amdgcn_wmma_f32_16x16x4_f32(
            /*neg_a=*/false, a, /*neg_b=*/false, ones,
            /*c_mod=*/(short)0, c, /*reuse_a=*/false, /*reuse_b=*/false);
    float t = ((c[0] + c[1]) + (c[2] + c[3])) + ((c[4] + c[5]) + (c[6] + c[7]));

    v2f a2; a2[0] = t; a2[1] = 0.0f;
    v8f d = {};
    d = __builtin_amdgcn_wmma_f32_16x16x4_f32(
            false, a2, false, ones, (short)0, d, false, false);
    return d[0];
}

// LUT {0,0,1,1,1,1,1,1,0,0} as a bitmask: bit i == lut[i]
#define LUT_MASK 0x0FCu

__global__ void __launch_bounds__(THREADS)
loss_partial_kernel(const float* __restrict__ pred,
                    const int*   __restrict__ target,
                    float*       __restrict__ partial,
                    int n) {
    float acc = 0.0f;
    const int stride = gridDim.x * blockDim.x;
    for (int row = blockIdx.x * blockDim.x + threadIdx.x; row < n; row += stride) {
        const float2* p = reinterpret_cast<const float2*>(pred + (size_t)row * 10);
        float2 v0 = p[0], v1 = p[1], v2 = p[2], v3 = p[3], v4 = p[4];
        float x[10];
        x[0] = v0.x; x[1] = v0.y; x[2] = v1.x; x[3] = v1.y; x[4] = v2.x;
        x[5] = v2.y; x[6] = v3.x; x[7] = v3.y; x[8] = v4.x; x[9] = v4.y;

        const int t = target[row];

        // max + first-occurrence argmax (matches jnp.argmax tie-break)
        float m = x[0]; int bi = 0;
        #pragma unroll
        for (int j = 1; j < 10; ++j) {
            if (x[j] > m) { m = x[j]; bi = j; }
        }

        // sum of exp(x - m)
        float s = 0.0f;
        #pragma unroll
        for (int j = 0; j < 10; ++j) s += __expf(x[j] - m);

        // x[target] without dynamic register indexing
        float xt = x[0];
        #pragma unroll
        for (int j = 1; j < 10; ++j) xt = (t == j) ? x[j] : xt;

        // CE term: logsumexp - x_t
        float ce = (m + __logf(s)) - xt;

        // BCE with {0,1} probs and -100 clamp == 100 * (b_pred != b_tgt)
        int bp = (LUT_MASK >> bi) & 1;
        int bt = (LUT_MASK >> t) & 1;
        float bce = (bp != bt) ? 100.0f : 0.0f;

        acc += ce + bce;
    }

    // wave32 reduction on the tensor path (uniform control flow: EXEC all-1s)
    float wsum = wave_reduce_wmma(acc);

    __shared__ float lds[WAVES_PER_BLOCK];
    const int wave = threadIdx.x >> 5;
    const int lane = threadIdx.x & 31;
    if (lane == 0) lds[wave] = wsum;
    __syncthreads();
    if (threadIdx.x == 0) {
        float b = 0.0f;
        #pragma unroll
        for (int w = 0; w < WAVES_PER_BLOCK; ++w) b += lds[w];
        partial[blockIdx.x] = b;
    }
}

__global__ void __launch_bounds__(THREADS)
loss_final_kernel(const float* __restrict__ partial,
                  float*       __restrict__ out,
                  int nblocks, float inv_n) {
    float acc = 0.0f;
    for (int i = threadIdx.x; i < nblocks; i += blockDim.x) acc += partial[i];

    float wsum = wave_reduce_wmma(acc);

    __shared__ float lds[WAVES_PER_BLOCK];
    const int wave = threadIdx.x >> 5;
    const int lane = threadIdx.x & 31;
    if (lane == 0) lds[wave] = wsum;
    __syncthreads();
    if (threadIdx.x == 0) {
        float b = 0.0f;
        #pragma unroll
        for (int w = 0; w < WAVES_PER_BLOCK; ++w) b += lds[w];
        out[0] = b * inv_n;
    }
}

extern "C" void kernel_launch(void* const* d_in, const int* in_sizes, int n_in,
                              void* d_out, int out_size, void* d_ws, size_t ws_size,
                              hipStream_t stream) {
    const float* pred   = (const float*)d_in[0];
    const int*   target = (const int*)d_in[1];
    float*       out    = (float*)d_out;
    float*       partial = (float*)d_ws;   // BLOCKS * 4 bytes = 4 KB
    const int n = in_sizes[1];             // N rows (target count)

    loss_partial_kernel<<<BLOCKS, THREADS, 0, stream>>>(pred, target, partial, n);
    loss_final_kernel<<<1, THREADS, 0, stream>>>(partial, out, BLOCKS, 1.0f / (float)n);
}